// GNN_23046794510938
// MI455X (gfx1250) — compile-verified
//
#include <hip/hip_runtime.h>
#include <hip/hip_bf16.h>

typedef float v2f __attribute__((ext_vector_type(2)));
typedef float v8f __attribute__((ext_vector_type(8)));

#define TPB 256

// ---------------- degree / norm precompute (shared by all 4 layers) --------

__global__ void k_init_deg(float* __restrict__ deg, int n) {
    int i = blockIdx.x * blockDim.x + threadIdx.x;
    if (i < n) deg[i] = 1.0f;                       // self-loop contribution
}

__global__ void k_acc_deg(const int* __restrict__ col, float* __restrict__ deg, int E) {
    int e = blockIdx.x * blockDim.x + threadIdx.x;
    if (e < E) atomicAdd(&deg[col[e]], 1.0f);
}

__global__ void k_dinv(float* __restrict__ deg, int n) {
    int i = blockIdx.x * blockDim.x + threadIdx.x;
    if (i < n) deg[i] = rsqrtf(deg[i]);             // in-place deg -> dinv
}

__global__ void k_edge_norm(const int* __restrict__ row, const int* __restrict__ col,
                            const float* __restrict__ dinv, float* __restrict__ wn, int E) {
    int e = blockIdx.x * blockDim.x + threadIdx.x;
    if (e < E) wn[e] = dinv[row[e]] * dinv[col[e]];
}

// ---------------- WMMA f32 16x16x4 node-feature transform ------------------
// H = act(X) @ W^T   (act = ReLU for layers 2..4)
// Epilogue also seeds OUT with bias + self-loop message: OUT = b + dinv^2 * H.
template <int CIN, int COUT, bool RELU>
__global__ void k_gemm_wmma(const float* __restrict__ X, const float* __restrict__ W,
                            const float* __restrict__ bias, const float* __restrict__ dinv,
                            float* __restrict__ H, float* __restrict__ OUT, int n_row_tiles) {
    constexpr int CT = COUT / 16;                   // column tiles
    int wave = (int)((blockIdx.x * blockDim.x + threadIdx.x) >> 5);
    int lane = threadIdx.x & 31;
    int rt = wave / CT;
    int ct = wave - rt * CT;
    if (rt >= n_row_tiles) return;                  // wave-uniform: EXEC stays all-1s

    const int m   = lane & 15;                      // A: row M; B: col N; D: col N
    const int kh  = lane >> 4;                      // K-half selector
    const int row0 = rt << 4;
    const int co0  = ct << 4;

    const float* xr = X + (size_t)(row0 + m) * CIN;
    const float* wr = W + (size_t)(co0 + m) * CIN;  // B[k][n] = W[co0+n][k]

    v8f acc = {};
#pragma unroll
    for (int k0 = 0; k0 < CIN; k0 += 4) {
        int ka = k0 + 2 * kh;
        v2f a, b;
        float ax = (ka     < CIN) ? xr[ka]     : 0.0f;
        float ay = (ka + 1 < CIN) ? xr[ka + 1] : 0.0f;
        if (RELU) { ax = fmaxf(ax, 0.0f); ay = fmaxf(ay, 0.0f); }
        a.x = ax; a.y = ay;
        b.x = (ka     < CIN) ? wr[ka]     : 0.0f;
        b.y = (ka + 1 < CIN) ? wr[ka + 1] : 0.0f;
        // (neg_a, A, neg_b, B, c_mod, C, reuse_a, reuse_b)
        acc = __builtin_amdgcn_wmma_f32_16x16x4_f32(false, a, false, b,
                                                    (short)0, acc, false, false);
    }

    float bj = bias[co0 + m];                       // m == N for C/D layout
#pragma unroll
    for (int r = 0; r < 8; ++r) {                   // D VGPR r -> M = r + 8*kh
        int rowi = row0 + r + 8 * kh;
        float h  = acc[r];
        size_t idx = (size_t)rowi * COUT + co0 + m;
        H[idx] = h;
        float di = dinv[rowi];
        OUT[idx] = bj + di * di * h;                // bias + self-loop message
    }
}

// ---------------- edge scatter-add (dominant cost) -------------------------
// C/2 consecutive threads per edge -> coalesced float2 gathers + f32 atomics.
template <int C>
__global__ void k_scatter(const int* __restrict__ row, const int* __restrict__ col,
                          const float* __restrict__ wn, const float* __restrict__ H,
                          float* __restrict__ OUT, int E) {
    constexpr unsigned TPE = C / 2;
    unsigned t = blockIdx.x * blockDim.x + threadIdx.x;
    unsigned total = (unsigned)E * TPE;
    if (t >= total) return;
    unsigned e = t / TPE;
    unsigned f = (t - e * TPE) * 2u;
    int r = row[e], c = col[e];
    float w = wn[e];
    float2 v = *(const float2*)(H + (size_t)r * C + f);
    atomicAdd(OUT + (size_t)c * C + f,     w * v.x);
    atomicAdd(OUT + (size_t)c * C + f + 1, w * v.y);
}

// ---------------- pooling + MLP head ---------------------------------------

__global__ void k_init_gmax(int* __restrict__ g, int n) {
    int i = blockIdx.x * blockDim.x + threadIdx.x;
    if (i < n) g[i] = 0;                            // == +0.0f; implements fused ReLU
}

__global__ void k_pool(const float* __restrict__ H, const int* __restrict__ batch,
                       int* __restrict__ gmax, int n) {      // n = N*64
    int t = blockIdx.x * blockDim.x + threadIdx.x;
    if (t >= n) return;
    int i = t >> 6, f = t & 63;
    float v = H[t];
    if (v > 0.0f)                                    // nonneg floats: int max == float max
        atomicMax(&gmax[(batch[i] << 6) + f], __float_as_int(v));
}

__global__ void k_mlp1(const int* __restrict__ gmax, const float* __restrict__ W,
                       const float* __restrict__ b, float* __restrict__ g1, int G) {
    int t = blockIdx.x * blockDim.x + threadIdx.x;
    if (t >= G * 64) return;
    int g = t >> 6, j = t & 63;
    float s = b[j];
#pragma unroll
    for (int k = 0; k < 64; ++k)
        s += __int_as_float(gmax[(g << 6) + k]) * W[j * 64 + k];
    g1[t] = fmaxf(s, 0.0f);
}

__global__ void k_mlp2(const float* __restrict__ g1, const float* __restrict__ W,
                       const float* __restrict__ b, float* __restrict__ out, int G) {
    int t = blockIdx.x * blockDim.x + threadIdx.x;
    if (t >= G * 10) return;
    int g = t / 10, j = t - g * 10;
    float s = b[j];
#pragma unroll
    for (int k = 0; k < 64; ++k)
        s += g1[g * 64 + k] * W[j * 64 + k];
    out[t] = s;
}

// ---------------- driver ----------------------------------------------------

static inline int cdiv_l(long a, long b) { return (int)((a + b - 1) / b); }

extern "C" void kernel_launch(void* const* d_in, const int* in_sizes, int n_in,
                              void* d_out, int out_size, void* d_ws, size_t ws_size,
                              hipStream_t stream) {
    const int N = 100000, E = 3200000, G = 256;

    const float* x   = (const float*)d_in[0];
    const int*   ei  = (const int*)d_in[1];
    const int*   row = ei;
    const int*   col = ei + E;
    const int*   bat = (const int*)d_in[2];
    // d_in[3] = num_graphs (256, constant)
    const float* W1 = (const float*)d_in[4];  const float* b1 = (const float*)d_in[5];
    const float* W2 = (const float*)d_in[6];  const float* b2 = (const float*)d_in[7];
    const float* W3 = (const float*)d_in[8];  const float* b3 = (const float*)d_in[9];
    const float* W4 = (const float*)d_in[10]; const float* b4 = (const float*)d_in[11];
    const float* L1w = (const float*)d_in[12]; const float* L1b = (const float*)d_in[13];
    const float* L2w = (const float*)d_in[14]; const float* L2b = (const float*)d_in[15];
    float* out = (float*)d_out;

    // workspace layout
    char* ws = (char*)d_ws;
    float* dinv = (float*)ws;                                    // N floats
    size_t off  = 1u << 19;                                      // 512 KB
    float* wn   = (float*)(ws + off);                            // E floats
    off += (((size_t)E * 4) + 255) & ~(size_t)255;
    size_t BUF = (((size_t)N * 64 * 4) + 255) & ~(size_t)255;    // 25.6 MB
    float* buf0 = (float*)(ws + off);                            // per-layer H
    float* buf1 = (float*)(ws + off + BUF);                      // ping
    float* buf2 = (float*)(ws + off + 2 * BUF);                  // pong
    int*   gmax = (int*)  (ws + off + 3 * BUF);                  // 256*64 ints
    float* g1   = (float*)(ws + off + 3 * BUF + 65536);          // 256*64 floats

    // degrees & symmetric norms — shared across all 4 layers
    k_init_deg <<<cdiv_l(N, TPB), TPB, 0, stream>>>(dinv, N);
    k_acc_deg  <<<cdiv_l(E, TPB), TPB, 0, stream>>>(col, dinv, E);
    k_dinv     <<<cdiv_l(N, TPB), TPB, 0, stream>>>(dinv, N);
    k_edge_norm<<<cdiv_l(E, TPB), TPB, 0, stream>>>(row, col, dinv, wn, E);

    const int nrt = N / 16;                                      // 6250, exact

    // layer 1: x[N,3] -> H buf0[N,16], OUT buf1
    k_gemm_wmma<3, 16, false><<<cdiv_l((long)nrt * 1 * 32, TPB), TPB, 0, stream>>>(
        x, W1, b1, dinv, buf0, buf1, nrt);
    k_scatter<16><<<cdiv_l((long)E * 8, TPB), TPB, 0, stream>>>(row, col, wn, buf0, buf1, E);

    // layer 2: buf1 -> H buf0[N,32], OUT buf2   (ReLU fused into A-load)
    k_gemm_wmma<16, 32, true><<<cdiv_l((long)nrt * 2 * 32, TPB), TPB, 0, stream>>>(
        buf1, W2, b2, dinv, buf0, buf2, nrt);
    k_scatter<32><<<cdiv_l((long)E * 16, TPB), TPB, 0, stream>>>(row, col, wn, buf0, buf2, E);

    // layer 3: buf2 -> H buf0[N,48], OUT buf1
    k_gemm_wmma<32, 48, true><<<cdiv_l((long)nrt * 3 * 32, TPB), TPB, 0, stream>>>(
        buf2, W3, b3, dinv, buf0, buf1, nrt);
    k_scatter<48><<<cdiv_l((long)E * 24, TPB), TPB, 0, stream>>>(row, col, wn, buf0, buf1, E);

    // layer 4: buf1 -> H buf0[N,64], OUT buf2
    k_gemm_wmma<48, 64, true><<<cdiv_l((long)nrt * 4 * 32, TPB), TPB, 0, stream>>>(
        buf1, W4, b4, dinv, buf0, buf2, nrt);
    k_scatter<64><<<cdiv_l((long)E * 32, TPB), TPB, 0, stream>>>(row, col, wn, buf0, buf2, E);

    // global max pool (ReLU fused via zero init) + MLP head
    k_init_gmax<<<cdiv_l(G * 64, TPB), TPB, 0, stream>>>(gmax, G * 64);
    k_pool     <<<cdiv_l((long)N * 64, TPB), TPB, 0, stream>>>(buf2, bat, gmax, N * 64);
    k_mlp1     <<<cdiv_l(G * 64, TPB), TPB, 0, stream>>>(gmax, L1w, L1b, g1, G);
    k_mlp2     <<<cdiv_l(G * 10, TPB), TPB, 0, stream>>>(g1, L2w, L2b, out, G);
}